// GCN_16655883174243
// MI455X (gfx1250) — compile-verified
//
#include <hip/hip_runtime.h>
#include <hip/hip_bf16.h>
#include <cstdint>
#include <cstddef>

#define FDIM 128  // F_IN == HID == 128

typedef float v2f __attribute__((ext_vector_type(2)));
typedef float v8f __attribute__((ext_vector_type(8)));

// ---------------- degree / normalization ----------------
__global__ __launch_bounds__(256) void k_deg_init(float* __restrict__ deg, int n) {
  int i = blockIdx.x * 256 + threadIdx.x;
  if (i < n) deg[i] = 1.0f;  // self-loop contribution
}

__global__ __launch_bounds__(256) void k_deg_accum(const int* __restrict__ col,
                                                   float* __restrict__ deg, int ne) {
  int e = blockIdx.x * 256 + threadIdx.x;
  if (e < ne) unsafeAtomicAdd(&deg[col[e]], 1.0f);
}

__global__ __launch_bounds__(256) void k_deg_finish(float* __restrict__ dis, int n) {
  int i = blockIdx.x * 256 + threadIdx.x;
  if (i < n) dis[i] = rsqrtf(dis[i]);  // deg >= 1 always (self-loops)
}

// ---------------- fp32 WMMA GEMM: H = act(X [+ bias]) @ W ----------------
// One block = 16 rows; 8 waves each own a 16-wide N tile; K swept in steps of 4.
// If bias != nullptr, applies relu(x + bias) while staging A into LDS (fuses the
// previous layer's bias+ReLU epilogue into this GEMM's prologue).
__global__ __launch_bounds__(256) void k_gemm(const float* __restrict__ X,
                                              const float* __restrict__ W,
                                              const float* __restrict__ bias,
                                              float* __restrict__ H, int n) {
  __shared__ float As[16 * FDIM];  // 8 KB A tile
  const int row0 = blockIdx.x * 16;
  const int tid  = threadIdx.x;
  const int wave = tid >> 5;
  const int lane = tid & 31;
  const bool full = (row0 + 16 <= n);  // wave-uniform fast path

  {  // stage A tile (16 x 128) into LDS, float4-vectorized (512 float4 / 256 threads)
    const float4* Xv = (const float4*)(X + (size_t)row0 * FDIM);
    const float4* Bv = (const float4*)bias;
    float4* Av = (float4*)As;
#pragma unroll
    for (int i = 0; i < 2; ++i) {
      int idx = tid + i * 256;  // 0..511 ; row = idx>>5, col4 = idx&31
      float4 v = make_float4(0.f, 0.f, 0.f, 0.f);
      if (full || (row0 + (idx >> 5) < n)) v = Xv[idx];
      if (bias) {
        float4 bb = Bv[idx & 31];
        v.x = fmaxf(v.x + bb.x, 0.f);
        v.y = fmaxf(v.y + bb.y, 0.f);
        v.z = fmaxf(v.z + bb.z, 0.f);
        v.w = fmaxf(v.w + bb.w, 0.f);
      }
      Av[idx] = v;
    }
  }
  __syncthreads();

  const int m     = lane & 15;   // M row (A) / N col (B,C)
  const int khalf = lane >> 4;   // lanes 16-31 hold K+2/K+3
  const int n0    = wave * 16;

  v8f c = {};
#pragma unroll 4
  for (int k = 0; k < FDIM; k += 4) {
    const int kb = k + khalf * 2;
    v2f a, b;
    a.x = As[m * FDIM + kb];                    // A[m][kb]
    a.y = As[m * FDIM + kb + 1];                // A[m][kb+1]
    b.x = W[(size_t)kb * FDIM + n0 + m];        // B[kb][n]
    b.y = W[(size_t)(kb + 1) * FDIM + n0 + m];  // B[kb+1][n]
    c = __builtin_amdgcn_wmma_f32_16x16x4_f32(false, a, false, b, (short)0, c,
                                              false, false);
  }

  // C/D layout: VGPR v -> M = v + 8*khalf, N = m
  float* Hp = H + (size_t)row0 * FDIM + n0;
  if (full) {
#pragma unroll
    for (int v = 0; v < 8; ++v) {
      int mm = v + khalf * 8;
      Hp[(size_t)mm * FDIM + m] = c[v];
    }
  } else {
#pragma unroll
    for (int v = 0; v < 8; ++v) {
      int mm = v + khalf * 8;
      if (row0 + mm < n) Hp[(size_t)mm * FDIM + m] = c[v];
    }
  }
}

// ---------------- aggregation ----------------
// agg[i] = dis[i]^2 * h[i]   (self-loop term, also zero-initializes accumulator)
__global__ __launch_bounds__(256) void k_selfloop_init(const float* __restrict__ h,
                                                       const float* __restrict__ dis,
                                                       float* __restrict__ agg, int n) {
  int i = blockIdx.x * 256 + threadIdx.x;  // one float4 each; n*32 total
  int node = i >> 5;
  if (node < n) {
    float w = dis[node];
    w *= w;
    float4 v = ((const float4*)h)[i];
    v.x *= w; v.y *= w; v.z *= w; v.w *= w;
    ((float4*)agg)[i] = v;
  }
}

// one wave per edge: agg[col] += dis[row]*dis[col] * h[row]
__global__ __launch_bounds__(256) void k_scatter(const float* __restrict__ h,
                                                 const float* __restrict__ dis,
                                                 const int* __restrict__ row,
                                                 const int* __restrict__ col,
                                                 float* __restrict__ agg, int ne) {
  int e = (int)((blockIdx.x * 256 + threadIdx.x) >> 5);
  int lane = threadIdx.x & 31;
  if (e >= ne) return;
  if (lane == 0 && e + 4096 < ne) {  // stream the edge lists ahead of use
    __builtin_prefetch(row + e + 4096, 0, 1);  // -> global_prefetch_b8
    __builtin_prefetch(col + e + 4096, 0, 1);
  }
  int r = row[e];
  int c = col[e];
  float w = dis[r] * dis[c];
  float4 v = ((const float4*)(h + (size_t)r * FDIM))[lane];
  float* ap = agg + (size_t)c * FDIM + lane * 4;
  unsafeAtomicAdd(ap + 0, w * v.x);
  unsafeAtomicAdd(ap + 1, w * v.y);
  unsafeAtomicAdd(ap + 2, w * v.z);
  unsafeAtomicAdd(ap + 3, w * v.w);
}

// ---------------- global mean pool ----------------
__global__ __launch_bounds__(256) void k_pool_init(float* __restrict__ sums,
                                                   float* __restrict__ cnts, int ng) {
  int i = blockIdx.x * 256 + threadIdx.x;
  if (i < ng * FDIM) sums[i] = 0.0f;
  if (i < ng) cnts[i] = 0.0f;
}

// applies the last layer's bias+ReLU on the fly: sums[g] += relu(h[node] + bias)
__global__ __launch_bounds__(256) void k_pool_accum(const float* __restrict__ h,
                                                    const float* __restrict__ bias,
                                                    const int* __restrict__ batch,
                                                    float* __restrict__ sums,
                                                    float* __restrict__ cnts, int n) {
  int node = (int)((blockIdx.x * 256 + threadIdx.x) >> 5);
  int lane = threadIdx.x & 31;
  if (node >= n) return;
  int g = batch[node];
  float4 v  = ((const float4*)(h + (size_t)node * FDIM))[lane];
  float4 bb = ((const float4*)bias)[lane];
  v.x = fmaxf(v.x + bb.x, 0.f);
  v.y = fmaxf(v.y + bb.y, 0.f);
  v.z = fmaxf(v.z + bb.z, 0.f);
  v.w = fmaxf(v.w + bb.w, 0.f);
  float* sp = sums + (size_t)g * FDIM + lane * 4;
  unsafeAtomicAdd(sp + 0, v.x);
  unsafeAtomicAdd(sp + 1, v.y);
  unsafeAtomicAdd(sp + 2, v.z);
  unsafeAtomicAdd(sp + 3, v.w);
  if (lane == 0) unsafeAtomicAdd(&cnts[g], 1.0f);
}

__global__ __launch_bounds__(256) void k_pool_finish(float* __restrict__ out,
                                                     const float* __restrict__ cnts,
                                                     int ng) {
  int i = blockIdx.x * 256 + threadIdx.x;
  if (i < ng * FDIM) out[i] = out[i] / fmaxf(cnts[i >> 7], 1.0f);
}

// ---------------- launcher ----------------
extern "C" void kernel_launch(void* const* d_in, const int* in_sizes, int n_in,
                              void* d_out, int out_size, void* d_ws, size_t ws_size,
                              hipStream_t stream) {
  (void)n_in; (void)ws_size;
  const float* x = (const float*)d_in[0];
  const float* Wm[4] = {(const float*)d_in[1], (const float*)d_in[3],
                        (const float*)d_in[5], (const float*)d_in[7]};
  const float* Bv[4] = {(const float*)d_in[2], (const float*)d_in[4],
                        (const float*)d_in[6], (const float*)d_in[8]};
  const int* ei    = (const int*)d_in[9];
  const int* batch = (const int*)d_in[10];

  const int n  = in_sizes[0] / FDIM;   // 100000 nodes
  const int ne = in_sizes[9] / 2;      // 1600000 edges
  const int ng = out_size / FDIM;      // 2048 graphs
  const int* row = ei;                 // edge_index[0] (source)
  const int* col = ei + ne;            // edge_index[1] (target)

  char* ws    = (char*)d_ws;
  float* dis  = (float*)ws;                                     // n floats
  float* cnts = (float*)(ws + ((size_t)1 << 19));               // ng floats
  float* bufA = (float*)(ws + ((size_t)1 << 21));               // n*128 floats (h)
  float* bufB = (float*)(ws + ((size_t)1 << 21) + ((size_t)54 << 20));  // agg
  float* outp = (float*)d_out;

  const int nblk  = (n + 255) / 256;
  const int eblk  = (ne + 255) / 256;
  const int f4blk = (n * 32 + 255) / 256;                       // n*128 floats as float4
  const int wvblk = (int)(((long long)ne * 32 + 255) / 256);    // 1 wave per edge
  const int pnblk = (n * 32 + 255) / 256;                       // 1 wave per node
  const int gblk  = (ng * FDIM + 255) / 256;

  // symmetric normalization coefficients: dis = rsqrt(1 + in-degree)
  k_deg_init  <<<nblk, 256, 0, stream>>>(dis, n);
  k_deg_accum <<<eblk, 256, 0, stream>>>(col, dis, ne);
  k_deg_finish<<<nblk, 256, 0, stream>>>(dis, n);

  // layer l: h = relu(prev_agg + b_{l-1}) @ W_l ; agg = D^-1/2 A D^-1/2 h
  const float* in = x;
  for (int l = 0; l < 4; ++l) {
    const float* bias = (l == 0) ? nullptr : Bv[l - 1];
    k_gemm         <<<(n + 15) / 16, 256, 0, stream>>>(in, Wm[l], bias, bufA, n);
    k_selfloop_init<<<f4blk, 256, 0, stream>>>(bufA, dis, bufB, n);
    k_scatter      <<<wvblk, 256, 0, stream>>>(bufA, dis, row, col, bufB, ne);
    in = bufB;
  }

  // global mean pool of relu(bufB + b4)
  k_pool_init  <<<gblk, 256, 0, stream>>>(outp, cnts, ng);
  k_pool_accum <<<pnblk, 256, 0, stream>>>(bufB, Bv[3], batch, outp, cnts, n);
  k_pool_finish<<<gblk, 256, 0, stream>>>(outp, cnts, ng);
}